// hash_encoding_78847009620517
// MI455X (gfx1250) — compile-verified
//
#include <hip/hip_runtime.h>
#include <cmath>

typedef float v2f __attribute__((ext_vector_type(2)));

#define NUM_LEVELS 16
#define LOG2_T 19

// One thread per (point, level). Block of 256 threads covers 16 points x 16 levels.
__global__ __launch_bounds__(256) void hashenc_kernel(
    const float* __restrict__ pts,
    const v2f*   __restrict__ tab,
    v2f*         __restrict__ out,
    const float* __restrict__ resbuf,
    int n_points)
{
    __shared__ float spts[48];                 // 16 points * 3 coords
    const int tid   = threadIdx.x;
    const int pbase = blockIdx.x * 16;

    // CDNA5 async DMA: stage 192B of point coords global->LDS (ASYNCcnt path).
    if (tid < 48) {
        const int gofs = pbase * 3 + tid;
        if (gofs < n_points * 3) {
            __builtin_amdgcn_global_load_async_to_lds_b32(
                (__attribute__((address_space(1))) int*)(pts + gofs),
                (__attribute__((address_space(3))) int*)(&spts[tid]),
                0, 0);
        }
    }
    __builtin_amdgcn_s_wait_asynccnt(0);
    __syncthreads();

    const int pl  = tid >> 4;                  // local point 0..15
    const int lvl = tid & 15;                  // level 0..15
    const int p   = pbase + pl;
    if (p >= n_points) return;

    const float r  = resbuf[lvl];
    const float sx = spts[pl * 3 + 0] * r;
    const float sy = spts[pl * 3 + 1] * r;
    const float sz = spts[pl * 3 + 2] * r;

    const float fx = floorf(sx), fy = floorf(sy), fz = floorf(sz);
    const float ox = sx - fx, oy = sy - fy, oz = sz - fz;

    // Coords are non-negative; int64 mul + mod 2^19 == uint32 mul + mask.
    const uint32_t xf = (uint32_t)(int)fx;
    const uint32_t yf = (uint32_t)(int)fy;
    const uint32_t zf = (uint32_t)(int)fz;
    const uint32_t xc = (uint32_t)(int)ceilf(sx);
    const uint32_t yc = (uint32_t)(int)ceilf(sy);
    const uint32_t zc = (uint32_t)(int)ceilf(sz);

    const uint32_t hyf = yf * 2654435761u, hyc = yc * 2654435761u;
    const uint32_t hzf = zf * 805459861u,  hzc = zc * 805459861u;
    const uint32_t M    = (1u << LOG2_T) - 1u;
    const uint32_t base = (uint32_t)lvl << LOG2_T;

    // Corner order per reference CORNER_MASK (1 = ceil):
    const uint32_t i0 = base + ((xc ^ hyc ^ hzc) & M); // (c,c,c)
    const uint32_t i1 = base + ((xc ^ hyf ^ hzc) & M); // (c,f,c)
    const uint32_t i2 = base + ((xf ^ hyf ^ hzc) & M); // (f,f,c)
    const uint32_t i3 = base + ((xf ^ hyc ^ hzc) & M); // (f,c,c)
    const uint32_t i4 = base + ((xc ^ hyc ^ hzf) & M); // (c,c,f)
    const uint32_t i5 = base + ((xc ^ hyf ^ hzf) & M); // (c,f,f)
    const uint32_t i6 = base + ((xf ^ hyf ^ hzf) & M); // (f,f,f)
    const uint32_t i7 = base + ((xf ^ hyc ^ hzf) & M); // (f,c,f)

    // 8 independent 8-byte gathers (global_load_b64), issued back-to-back.
    const v2f f0 = tab[i0];
    const v2f f1 = tab[i1];
    const v2f f2 = tab[i2];
    const v2f f3 = tab[i3];
    const v2f f4 = tab[i4];
    const v2f f5 = tab[i5];
    const v2f f6 = tab[i6];
    const v2f f7 = tab[i7];

    const float ox1 = 1.0f - ox, oy1 = 1.0f - oy, oz1 = 1.0f - oz;
    const v2f f03   = f0 * ox + f3 * ox1;
    const v2f f12   = f1 * ox + f2 * ox1;
    const v2f f56   = f5 * ox + f6 * ox1;
    const v2f f47   = f4 * ox + f7 * ox1;
    const v2f f0312 = f03 * oy + f12 * oy1;
    const v2f f4756 = f47 * oy + f56 * oy1;
    const v2f enc   = f0312 * oz + f4756 * oz1;

    // Streaming output, never re-read: non-temporal so L2 stays for the table.
    __builtin_nontemporal_store(enc, out + (p * NUM_LEVELS + lvl));
}

extern "C" void kernel_launch(void* const* d_in, const int* in_sizes, int n_in,
                              void* d_out, int out_size, void* d_ws, size_t ws_size,
                              hipStream_t stream)
{
    const float* pts = (const float*)d_in[0];
    const v2f*   tab = (const v2f*)d_in[1];
    v2f*         out = (v2f*)d_out;
    const int n_points = in_sizes[0] / 3;

    // Replicate the reference's float64 resolution table exactly on the host:
    // b = exp((log(2048)-log(16))/15); res[i] = floor(16 * b^i).
    // (Several levels are within ~5e-3 of an integer -> must use fp64 libm.)
    static float res_host[NUM_LEVELS];
    const double b = std::exp((std::log(2048.0) - std::log(16.0)) / 15.0);
    for (int i = 0; i < NUM_LEVELS; ++i)
        res_host[i] = (float)std::floor(16.0 * std::pow(b, (double)i));
    (void)hipMemcpyAsync(d_ws, res_host, sizeof(res_host), hipMemcpyHostToDevice, stream);

    const int total = n_points * NUM_LEVELS;
    const int block = 256;
    const int grid  = (total + block - 1) / block;
    hashenc_kernel<<<grid, block, 0, stream>>>(pts, tab, out, (const float*)d_ws, n_points);
}